// DeltaModel_3204045603621
// MI455X (gfx1250) — compile-verified
//
#include <hip/hip_runtime.h>

// ---------------------------------------------------------------------------
// B=64, L=4096, H=64, VOCAB=64.
// Structure: encoder collapses to a 64x64 token table (VOCAB distinct h's),
// 1/(k.k+eps) collapses to a 64-entry table. Delta-rule scan runs with M in
// registers: 2 threads per row (half-row each as f32x2 pairs -> v_pk_fma_f32),
// k broadcast from LDS, v_pred reduced with one __shfl_xor per step.
// GEMM-shaped work uses fp32 WMMA (v_wmma_f32_16x16x4_f32).
// ---------------------------------------------------------------------------

#define HDIM   64
#define VOCABN 64
#define BATCH  64
#define SEQL   4096

typedef __attribute__((ext_vector_type(2))) float v2f;
typedef __attribute__((ext_vector_type(2))) float f32x2;
typedef __attribute__((ext_vector_type(8))) float v8f;

// --- WMMA f32 16x16x4 fragment helpers (ISA 7.12.2 layouts) -----------------
// A (16x4): lanes 0-15 -> M=lane, K={0,1}; lanes 16-31 -> M=lane-16, K={2,3}
__device__ __forceinline__ v2f frag_a(const float* A, int lda, int mt, int kt, int lane) {
  int row = (mt << 4) + (lane & 15);
  int k0  = (kt << 2) + ((lane >> 4) << 1);
  v2f a;
  a.x = A[row * lda + k0];
  a.y = A[row * lda + k0 + 1];
  return a;
}
// B (4x16): lanes 0-15 -> N=lane, K={0,1}; lanes 16-31 -> N=lane-16, K={2,3}
__device__ __forceinline__ v2f frag_b(const float* Bm, int ldb, int kt, int nt, int lane) {
  int col = (nt << 4) + (lane & 15);
  int k0  = (kt << 2) + ((lane >> 4) << 1);
  v2f b;
  b.x = Bm[k0 * ldb + col];
  b.y = Bm[(k0 + 1) * ldb + col];
  return b;
}
// C/D (16x16): VGPR v, lanes 0-15 -> M=v, N=lane; lanes 16-31 -> M=v+8.

__device__ __forceinline__ v8f wmma_f32(v2f a, v2f b, v8f c) {
  return __builtin_amdgcn_wmma_f32_16x16x4_f32(false, a, false, b, (short)0, c,
                                               false, false);
}

// ---------------------------------------------------------------------------
// Kernel 1: token table. T[v] = LN(e_v + relu(e_v@W1+b1)@W2+b2)*gamma+beta,
// rdenom[v] = 1/(T[v].T[v] + 1e-6). One workgroup, 8 waves, fp32 WMMA.
// ---------------------------------------------------------------------------
__global__ __launch_bounds__(256) void encode_table_kernel(
    const float* __restrict__ embed, const float* __restrict__ W1,
    const float* __restrict__ b1, const float* __restrict__ W2,
    const float* __restrict__ b2, const float* __restrict__ gamma,
    const float* __restrict__ beta, float* __restrict__ Tout,
    float* __restrict__ rdenom) {
  __shared__ float Rlds[64 * 128];  // relu(E@W1+b1)
  __shared__ float Xlds[64 * 64];   // E + R@W2 + b2
  const int tid  = threadIdx.x;
  const int wave = tid >> 5;
  const int lane = tid & 31;

  // GEMM1: R = relu(E @ W1 + b1)   (64x64 @ 64x128)
  for (int tile = wave; tile < 32; tile += 8) {
    int mt = tile >> 3, nt = tile & 7;
    v8f acc = {};
#pragma unroll
    for (int kt = 0; kt < 16; ++kt) {
      v2f a = frag_a(embed, 64, mt, kt, lane);
      v2f b = frag_b(W1, 128, kt, nt, lane);
      acc = wmma_f32(a, b, acc);
    }
    int col   = (nt << 4) + (lane & 15);
    int rbase = (mt << 4) + ((lane >> 4) << 3);
#pragma unroll
    for (int v = 0; v < 8; ++v) {
      float x = acc[v] + b1[col];
      Rlds[(rbase + v) * 128 + col] = fmaxf(x, 0.0f);
    }
  }
  __syncthreads();

  // GEMM2: X = E + R @ W2 + b2     (64x128 @ 128x64)
  for (int tile = wave; tile < 16; tile += 8) {
    int mt = tile >> 2, nt = tile & 3;
    v8f acc = {};
#pragma unroll
    for (int kt = 0; kt < 32; ++kt) {
      v2f a = frag_a(Rlds, 128, mt, kt, lane);
      v2f b = frag_b(W2, 64, kt, nt, lane);
      acc = wmma_f32(a, b, acc);
    }
    int col   = (nt << 4) + (lane & 15);
    int rbase = (mt << 4) + ((lane >> 4) << 3);
#pragma unroll
    for (int v = 0; v < 8; ++v) {
      int row = rbase + v;
      Xlds[row * 64 + col] = embed[row * 64 + col] + acc[v] + b2[col];
    }
  }
  __syncthreads();

  // LayerNorm each vocab row (population variance, eps=1e-5) + 1/(k.k+1e-6)
  if (tid < 64) {
    const float* xr = &Xlds[tid * 64];
    float mu = 0.f;
#pragma unroll
    for (int c = 0; c < 64; ++c) mu += xr[c];
    mu *= (1.0f / 64.0f);
    float var = 0.f;
#pragma unroll
    for (int c = 0; c < 64; ++c) { float d = xr[c] - mu; var += d * d; }
    var *= (1.0f / 64.0f);
    float rs  = rsqrtf(var + 1e-5f);
    float ksq = 0.f;
#pragma unroll
    for (int c = 0; c < 64; ++c) {
      float tv = (xr[c] - mu) * rs * gamma[c] + beta[c];
      Tout[tid * 64 + c] = tv;
      ksq += tv * tv;
    }
    rdenom[tid] = 1.0f / (ksq + 1e-6f);
  }
}

// ---------------------------------------------------------------------------
// Kernel 2: delta-rule scan. One block per batch, 128 threads (4 waves);
// thread pair (2r, 2r+1) owns M row r, each thread half a row as 16 f32x2
// (-> v_pk_fma_f32). Per step per thread: 8 ds_load_b128 + 32 pk_fma +
// 1 shfl_xor reduction. Zero global traffic inside the 4095-step loop.
// ---------------------------------------------------------------------------
__global__ __launch_bounds__(128) void delta_scan_kernel(
    const int* __restrict__ seq, const float* __restrict__ Ttab,
    const float* __restrict__ rdenom, float* __restrict__ ctx) {
  __shared__ float4 Tl4[VOCABN * 16];  // token table, 16B-aligned rows
  __shared__ float  rd[VOCABN];
  __shared__ int    toks[SEQL];
  const int b    = blockIdx.x;
  const int tid  = threadIdx.x;
  const int r    = tid >> 1;  // matrix row 0..63
  const int half = tid & 1;   // which half of the row's columns

  const float4* Tg4 = (const float4*)Ttab;
  for (int i = tid; i < VOCABN * 16; i += 128) Tl4[i] = Tg4[i];
  if (tid < VOCABN) rd[tid] = rdenom[tid];
  const int4* sg = (const int4*)(seq + b * SEQL);
  int4*       tl = (int4*)toks;
  for (int i = tid; i < SEQL / 4; i += 128) tl[i] = sg[i];
  __syncthreads();

  const float* Tl    = (const float*)Tl4;
  const int    kbase = half << 3;  // float4 offset within a table row

  f32x2 M2[16];  // this thread's half of M row r (32 columns)
#pragma unroll
  for (int i = 0; i < 16; ++i) M2[i] = (f32x2){0.f, 0.f};

  for (int t = 0; t < SEQL - 1; ++t) {
    int tok = toks[t];
    const float4* kp = &Tl4[tok * 16 + kbase];
    f32x2 kv[16];
    f32x2 acc0 = (f32x2){0.f, 0.f};
    f32x2 acc1 = (f32x2){0.f, 0.f};
#pragma unroll
    for (int i = 0; i < 8; ++i) {
      float4 ld = kp[i];
      kv[2 * i]     = (f32x2){ld.x, ld.y};
      kv[2 * i + 1] = (f32x2){ld.z, ld.w};
      acc0 = __builtin_elementwise_fma(M2[2 * i], kv[2 * i], acc0);
      acc1 = __builtin_elementwise_fma(M2[2 * i + 1], kv[2 * i + 1], acc1);
    }
    float part = (acc0.x + acc0.y) + (acc1.x + acc1.y);
    float vp   = part + __shfl_xor(part, 1, 32);   // sum the two half-rows
    float dv   = Tl[tok * 64 + r] - vp * rd[tok];  // k[r] - v_pred[r]/denom
    f32x2 dvv  = (f32x2){dv, dv};
#pragma unroll
    for (int i = 0; i < 16; ++i)
      M2[i] = __builtin_elementwise_fma(dvv, kv[i], M2[i]);
  }

  // ctx[b] = M q,  q = T[seq[b, L-1]]
  int tok = toks[SEQL - 1];
  const float4* qp = &Tl4[tok * 16 + kbase];
  f32x2 acc0 = (f32x2){0.f, 0.f};
  f32x2 acc1 = (f32x2){0.f, 0.f};
#pragma unroll
  for (int i = 0; i < 8; ++i) {
    float4 ld = qp[i];
    acc0 = __builtin_elementwise_fma(M2[2 * i], (f32x2){ld.x, ld.y}, acc0);
    acc1 = __builtin_elementwise_fma(M2[2 * i + 1], (f32x2){ld.z, ld.w}, acc1);
  }
  float part = (acc0.x + acc0.y) + (acc1.x + acc1.y);
  float vp   = part + __shfl_xor(part, 1, 32);
  if (half == 0) ctx[b * 64 + r] = vp;
}

// ---------------------------------------------------------------------------
// Kernel 3: out = (ctx @ Wr + br) @ Wo + bo.  Two 64x64x64 fp32 WMMA GEMMs.
// ---------------------------------------------------------------------------
__global__ __launch_bounds__(256) void head_kernel(
    const float* __restrict__ ctx, const float* __restrict__ Wr,
    const float* __restrict__ br, const float* __restrict__ Wo,
    const float* __restrict__ bo, float* __restrict__ out) {
  __shared__ float Glds[64 * 64];
  const int tid  = threadIdx.x;
  const int wave = tid >> 5;
  const int lane = tid & 31;

  for (int tile = wave; tile < 16; tile += 8) {
    int mt = tile >> 2, nt = tile & 3;
    v8f acc = {};
#pragma unroll
    for (int kt = 0; kt < 16; ++kt) {
      v2f a = frag_a(ctx, 64, mt, kt, lane);
      v2f b = frag_b(Wr, 64, kt, nt, lane);
      acc = wmma_f32(a, b, acc);
    }
    int col   = (nt << 4) + (lane & 15);
    int rbase = (mt << 4) + ((lane >> 4) << 3);
#pragma unroll
    for (int v = 0; v < 8; ++v) Glds[(rbase + v) * 64 + col] = acc[v] + br[col];
  }
  __syncthreads();

  for (int tile = wave; tile < 16; tile += 8) {
    int mt = tile >> 2, nt = tile & 3;
    v8f acc = {};
#pragma unroll
    for (int kt = 0; kt < 16; ++kt) {
      v2f a = frag_a(Glds, 64, mt, kt, lane);
      v2f b = frag_b(Wo, 64, kt, nt, lane);
      acc = wmma_f32(a, b, acc);
    }
    int col   = (nt << 4) + (lane & 15);
    int rbase = (mt << 4) + ((lane >> 4) << 3);
#pragma unroll
    for (int v = 0; v < 8; ++v) out[(rbase + v) * 64 + col] = acc[v] + bo[col];
  }
}

// ---------------------------------------------------------------------------
extern "C" void kernel_launch(void* const* d_in, const int* in_sizes, int n_in,
                              void* d_out, int out_size, void* d_ws, size_t ws_size,
                              hipStream_t stream) {
  const int*   seq   = (const int*)d_in[0];
  const float* embed = (const float*)d_in[1];
  const float* W1    = (const float*)d_in[2];
  const float* b1    = (const float*)d_in[3];
  const float* W2    = (const float*)d_in[4];
  const float* b2    = (const float*)d_in[5];
  const float* gamma = (const float*)d_in[6];
  const float* beta  = (const float*)d_in[7];
  const float* Wr    = (const float*)d_in[8];
  const float* br    = (const float*)d_in[9];
  const float* Wo    = (const float*)d_in[10];
  const float* bo    = (const float*)d_in[11];

  float* Tt     = (float*)d_ws;        // 64*64 token table
  float* rdenom = Tt + VOCABN * HDIM;  // 64
  float* ctx    = rdenom + VOCABN;     // 64*64
  float* out    = (float*)d_out;       // 64*64

  encode_table_kernel<<<1, 256, 0, stream>>>(embed, W1, b1, W2, b2, gamma, beta,
                                             Tt, rdenom);
  delta_scan_kernel<<<BATCH, 128, 0, stream>>>(seq, Tt, rdenom, ctx);
  head_kernel<<<1, 256, 0, stream>>>(ctx, Wr, br, Wo, bo, out);
}